// CSPN_38594576122096
// MI455X (gfx1250) — compile-verified
//
#include <hip/hip_runtime.h>

// CSPN: 24 launches of a 3x3 per-pixel-weighted stencil.
// Strategy: guidance (67MB) stays resident in the 192MB L2 across iterations;
// each iteration streams guidance from L2 and stages the h halo tile into LDS
// via CDNA5 async global->LDS loads (ASYNCcnt / s_wait_asynccnt).

#define TW 32   // tile width  (output)
#define THH 32  // tile height (output)
#define BDX 32
#define BDY 8

static constexpr int Bb = 8, Hh = 512, Ww = 512;
static constexpr int ITERS = 24;

typedef __attribute__((address_space(1))) int glb_i;   // global (AS1)
typedef __attribute__((address_space(3))) int lds_i;   // LDS (AS3)

__global__ __launch_bounds__(BDX * BDY) void cspn_step_kernel(
    const float* __restrict__ guid,   // [B,8,H,W]
    const float* __restrict__ hin,    // [B,1,H,W]
    float* __restrict__ hout)         // [B,1,H,W]
{
    __shared__ float tile[THH + 2][TW + 4];   // +4 cols: bank-conflict pad

    const int tx = threadIdx.x;
    const int ty = threadIdx.y;
    const int tid = ty * BDX + tx;
    const int bx = blockIdx.x * TW;
    const int by = blockIdx.y * THH;
    const int b  = blockIdx.z;

    const float* hplane = hin + (size_t)b * Hh * Ww;

    // ---- Stage (THH+2)x(TW+2) halo tile of h into LDS (clamped addresses,
    // always valid; zero-padding is applied as a mask at compute time). ----
    for (int i = tid; i < (THH + 2) * (TW + 2); i += BDX * BDY) {
        const int lr = i / (TW + 2);
        const int lc = i % (TW + 2);
        int gy = by + lr - 1; gy = gy < 0 ? 0 : (gy > Hh - 1 ? Hh - 1 : gy);
        int gx = bx + lc - 1; gx = gx < 0 ? 0 : (gx > Ww - 1 ? Ww - 1 : gx);
#if defined(__gfx1250__) && __has_builtin(__builtin_amdgcn_global_load_async_to_lds_b32) && __has_builtin(__builtin_amdgcn_s_wait_asynccnt)
        __builtin_amdgcn_global_load_async_to_lds_b32(
            (glb_i*)(hplane + (size_t)gy * Ww + gx),
            (lds_i*)&tile[lr][lc],
            /*imm offset=*/0, /*cpol=*/0);
#else
        tile[lr][lc] = hplane[(size_t)gy * Ww + gx];
#endif
    }
#if defined(__gfx1250__) && __has_builtin(__builtin_amdgcn_global_load_async_to_lds_b32) && __has_builtin(__builtin_amdgcn_s_wait_asynccnt)
    __builtin_amdgcn_s_wait_asynccnt(0);
#endif
    __syncthreads();

    // ---- Compute: h_new = h + s * sum_i guid_i * (n_i - h),
    //      s = 1 / max(sum_i |guid_i|, 1e-8), OOB neighbors masked to 0. ----
    const size_t gbase = (size_t)b * 8 * Hh * Ww;
    const size_t obase = (size_t)b * Hh * Ww;

#pragma unroll
    for (int k = 0; k < THH / BDY; ++k) {
        const int ry = ty + k * BDY;          // row within tile: 0..31
        const int y = by + ry;
        const int x = bx + tx;

        const float h = tile[ry + 1][tx + 1];
        float acc = 0.0f;
        float asum = 0.0f;

        int oi = 0;
#pragma unroll
        for (int di = -1; di <= 1; ++di) {
#pragma unroll
            for (int dj = -1; dj <= 1; ++dj) {
                if (di == 0 && dj == 0) continue;
                const float gr =
                    guid[gbase + (size_t)oi * Hh * Ww + (size_t)y * Ww + x];
                const int ny = y + di;
                const int nx = x + dj;
                const bool valid =
                    ((unsigned)ny < (unsigned)Hh) && ((unsigned)nx < (unsigned)Ww);
                const float n = valid ? tile[ry + 1 + di][tx + 1 + dj] : 0.0f;
                acc = fmaf(gr, n - h, acc);
                asum += fabsf(gr);
                ++oi;
            }
        }

        const float s = 1.0f / fmaxf(asum, 1e-8f);
        hout[obase + (size_t)y * Ww + x] = fmaf(s, acc, h);
    }
}

extern "C" void kernel_launch(void* const* d_in, const int* in_sizes, int n_in,
                              void* d_out, int out_size, void* d_ws, size_t ws_size,
                              hipStream_t stream) {
    const float* guid = (const float*)d_in[0];   // [8,8,512,512]
    const float* x    = (const float*)d_in[1];   // [8,1,512,512]
    float* out = (float*)d_out;
    float* ws  = (float*)d_ws;                   // needs 8*512*512*4 = 8 MiB

    dim3 grid(Ww / TW, Hh / THH, Bb);
    dim3 block(BDX, BDY, 1);

    const float* cur = x;
    for (int t = 0; t < ITERS; ++t) {
        float* dst = (t & 1) ? out : ws;         // t=23 (last, odd) -> d_out
        cspn_step_kernel<<<grid, block, 0, stream>>>(guid, cur, dst);
        cur = dst;
    }
}